// DynamicBaseLine_9852654977123
// MI455X (gfx1250) — compile-verified
//
#include <hip/hip_runtime.h>

// Causal sliding-window min, k = 64, rows of 131072 f32, 256 rows.
// Memory-bound (256 MiB traffic -> ~11us at 23.3 TB/s). Van Herk two-scan
// algorithm: per 64-block prefix/suffix mins in LDS, out = min(P[l], S[l-63]).
// Staging uses gfx1250 async global->LDS DMA (ASYNCcnt) when available.
// 2D grid: blockIdx.x = tile-in-row, blockIdx.y = row (no integer division).

#define NTHREADS 256
#define TILE     4096
#define HALO     64                       // = k
#define REGN     (TILE + HALO)            // 4160 staged elements
#define NBLK     (REGN / 64)              // 65 blocks of 64
// pad 4 dwords per 64-block: keeps 16B alignment, spreads scan accesses
#define PIDX(l)  ((l) + (((l) >> 6) << 2))
#define LDSN     (REGN + 4 * NBLK)        // 4420 floats per array

#if __has_builtin(__builtin_amdgcn_global_load_async_to_lds_b128) && \
    __has_builtin(__builtin_amdgcn_s_wait_asynccnt)
#define USE_ASYNC 1
#else
#define USE_ASYNC 0
#endif

typedef int v4i_cdna5 __attribute__((__vector_size__(4 * sizeof(int))));

__device__ __forceinline__ void stage16B(const float* __restrict__ g, float* l) {
#if USE_ASYNC
  // async DMA: global (as1) -> LDS (as3), tracked by ASYNCcnt.
  __builtin_amdgcn_global_load_async_to_lds_b128(
      (__attribute__((address_space(1))) v4i_cdna5*)g,
      (__attribute__((address_space(3))) v4i_cdna5*)l,
      /*offset=*/0, /*cpol=*/0);
#else
  *reinterpret_cast<float4*>(l) = *reinterpret_cast<const float4*>(g);
#endif
}

__global__ __launch_bounds__(NTHREADS)
void swmin64_kernel(const float* __restrict__ x, float* __restrict__ out,
                    int cols)
{
    __shared__ float xs[LDSN];   // staged input (padded layout)
    __shared__ float ps[LDSN];   // per-64-block prefix mins
    __shared__ float ss[LDSN];   // per-64-block suffix mins

    const int tid      = threadIdx.x;
    const int tile     = blockIdx.x;               // tile within row
    const int row      = blockIdx.y;               // row
    const long long rowBase  = (long long)row * (long long)cols;
    const int tileBase = tile * TILE;

    // ---- stage [tileBase-HALO, tileBase+TILE) of this row into LDS ----
    if (tile == 0) {
        if (tid < HALO) xs[PIDX(tid)] = __builtin_inff();   // +inf = min identity
        const float* src = x + rowBase;
        #pragma unroll
        for (int it = 0; it < TILE / 4 / NTHREADS; ++it) {
            const int v = it * NTHREADS + tid;              // vec4 index
            stage16B(src + v * 4, &xs[PIDX(HALO + v * 4)]);
        }
    } else {
        const float* src = x + rowBase + tileBase - HALO;
        #pragma unroll
        for (int it = 0; it < (REGN / 4 + NTHREADS - 1) / NTHREADS; ++it) {
            const int v = it * NTHREADS + tid;
            if (v < REGN / 4)
                stage16B(src + v * 4, &xs[PIDX(v * 4)]);
        }
    }
#if USE_ASYNC
    __builtin_amdgcn_s_wait_asynccnt(0);
#endif
    __syncthreads();

    // ---- van Herk scans: block size 64 == window size ----
    if (tid < NBLK) {                       // suffix scans (backward)
        const int lb = tid * 64;
        float r = __builtin_inff();
        #pragma unroll 8
        for (int o = 63; o >= 0; --o) {
            r = fminf(r, xs[PIDX(lb + o)]);
            ss[PIDX(lb + o)] = r;
        }
    } else if (tid >= 128 && tid < 128 + NBLK) {   // prefix scans (forward)
        const int lb = (tid - 128) * 64;
        float r = __builtin_inff();
        #pragma unroll 8
        for (int o = 0; o < 64; ++o) {
            r = fminf(r, xs[PIDX(lb + o)]);
            ps[PIDX(lb + o)] = r;
        }
    }
    __syncthreads();

    // ---- combine + wide store: out[l] = min(P[l], S[l-63]) ----
    float* dst = out + rowBase + tileBase;
    #pragma unroll
    for (int it = 0; it < TILE / 4 / NTHREADS; ++it) {
        const int v  = it * NTHREADS + tid;
        const int l0 = HALO + v * 4;
        const float4 p = *reinterpret_cast<const float4*>(&ps[PIDX(l0)]);
        float4 r;
        r.x = fminf(p.x, ss[PIDX(l0 - 63)]);
        r.y = fminf(p.y, ss[PIDX(l0 - 62)]);
        r.z = fminf(p.z, ss[PIDX(l0 - 61)]);
        r.w = fminf(p.w, ss[PIDX(l0 - 60)]);
        reinterpret_cast<float4*>(dst)[v] = r;
    }
}

extern "C" void kernel_launch(void* const* d_in, const int* in_sizes, int n_in,
                              void* d_out, int out_size, void* d_ws, size_t ws_size,
                              hipStream_t stream) {
    const float* x = (const float*)d_in[0];
    float* out = (float*)d_out;
    (void)d_ws; (void)ws_size; (void)n_in; (void)out_size;

    const int total = in_sizes[0];           // 256 * 131072
    int cols = 131072;                       // per reference setup
    if (total < cols || (total % cols) != 0) cols = total;  // defensive
    const int rows = total / cols;
    const int tilesPerRow = cols / TILE;     // 32

    dim3 grid(tilesPerRow, rows);            // 32 x 256 workgroups
    dim3 block(NTHREADS);
    swmin64_kernel<<<grid, block, 0, stream>>>(x, out, cols);
}